// LightGCNReg_32581621907918
// MI455X (gfx1250) — compile-verified
//
#include <hip/hip_runtime.h>
#include <stdint.h>

typedef float v4f __attribute__((ext_vector_type(4)));

#define TPB 256   // threads per block (8 waves of 32)
#define EPB 256   // edges staged per block in the SpMM kernel

// ---------------------------------------------------------------------------
// init: cur = out = concat(user_emb, item_emb); next = 0
// ---------------------------------------------------------------------------
__global__ __launch_bounds__(TPB) void lgcn_init(
    const float* __restrict__ uemb, const float* __restrict__ iemb,
    long long nu4, long long n4,
    float* __restrict__ cur, float* __restrict__ out, float* __restrict__ nxt) {
  long long i = (long long)blockIdx.x * TPB + threadIdx.x;
  if (i >= n4) return;
  v4f v = (i < nu4) ? ((const v4f*)uemb)[i] : ((const v4f*)iemb)[i - nu4];
  ((v4f*)cur)[i] = v;
  ((v4f*)out)[i] = v;
  v4f z = {0.f, 0.f, 0.f, 0.f};
  ((v4f*)nxt)[i] = z;
}

// ---------------------------------------------------------------------------
// COO SpMM: nxt[row,:] += val * cur[col,:]
//  - edge triples staged into LDS with async global->LDS loads (ASYNCcnt)
//  - 16 lanes per edge, float4 per lane (one 64-dim row per half-wave)
//  - scatter via f32 L2 atomics (no-return -> STOREcnt)
// ---------------------------------------------------------------------------
__global__ __launch_bounds__(TPB) void lgcn_spmm(
    const int* __restrict__ rows, const int* __restrict__ cols,
    const float* __restrict__ vals, long long nnz,
    const float* __restrict__ cur, float* __restrict__ nxt) {
  __shared__ int   s_rows[EPB];
  __shared__ int   s_cols[EPB];
  __shared__ float s_vals[EPB];

  const int t = threadIdx.x;
  const long long base = (long long)blockIdx.x * EPB;
  long long idx = base + t;
  if (idx >= nnz) idx = nnz - 1;   // clamp keeps the async load in-bounds

  // LDS byte offsets: low 32 bits of the generic (flat) address of a
  // __shared__ object are the workgroup-relative LDS offset.
  unsigned dr = (unsigned)(uintptr_t)&s_rows[t];
  unsigned dc = (unsigned)(uintptr_t)&s_cols[t];
  unsigned dv = (unsigned)(uintptr_t)&s_vals[t];
  uint64_t ar = (uint64_t)(uintptr_t)(rows + idx);
  uint64_t ac = (uint64_t)(uintptr_t)(cols + idx);
  uint64_t av = (uint64_t)(uintptr_t)(vals + idx);

  asm volatile("global_load_async_to_lds_b32 %0, %1, off"
               :: "v"(dr), "v"(ar) : "memory");
  asm volatile("global_load_async_to_lds_b32 %0, %1, off"
               :: "v"(dc), "v"(ac) : "memory");
  asm volatile("global_load_async_to_lds_b32 %0, %1, off"
               :: "v"(dv), "v"(av) : "memory");
  asm volatile("s_wait_asynccnt 0x0" ::: "memory");
  __syncthreads();

  const int grp = t >> 4;   // 16 groups of 16 lanes
  const int l16 = t & 15;   // lane within group: owns dims [4*l16, 4*l16+3]

  for (int j = 0; j < EPB / 16; ++j) {
    const int e = grp * (EPB / 16) + j;
    if (base + e >= nnz) break;          // tail guard (monotone within group)
    const int   r = s_rows[e];           // LDS broadcast within the group
    const int   c = s_cols[e];
    const float w = s_vals[e];
    v4f v = ((const v4f*)cur)[(long long)c * 16 + l16];  // 256B coalesced gather
    float* dst = nxt + (long long)r * 64 + l16 * 4;
    atomicAdd(dst + 0, w * v.x);
    atomicAdd(dst + 1, w * v.y);
    atomicAdd(dst + 2, w * v.z);
    atomicAdd(dst + 3, w * v.w);
  }
}

// ---------------------------------------------------------------------------
// out += layer_result; zero the buffer that becomes the next scatter target.
// `out` is streamed non-temporally so the 3 sequential passes don't evict the
// gather-hot embedding table (which we want resident in the 192MB L2).
// ---------------------------------------------------------------------------
__global__ __launch_bounds__(TPB) void lgcn_add_zero(
    float* __restrict__ out, const float* __restrict__ nxt,
    float* __restrict__ zbuf, long long n4) {
  long long i = (long long)blockIdx.x * TPB + threadIdx.x;
  if (i >= n4) return;
  v4f a = __builtin_nontemporal_load((const v4f*)out + i);
  v4f b = ((const v4f*)nxt)[i];          // stays cached: it is next layer's cur
  __builtin_nontemporal_store(a + b, (v4f*)out + i);
  v4f z = {0.f, 0.f, 0.f, 0.f};
  ((v4f*)zbuf)[i] = z;
}

// Final layer: out = (out + nxt) * 0.25
__global__ __launch_bounds__(TPB) void lgcn_add_scale(
    float* __restrict__ out, const float* __restrict__ nxt, long long n4) {
  long long i = (long long)blockIdx.x * TPB + threadIdx.x;
  if (i >= n4) return;
  v4f a = __builtin_nontemporal_load((const v4f*)out + i);
  v4f b = ((const v4f*)nxt)[i];
  v4f r = (a + b) * 0.25f;
  __builtin_nontemporal_store(r, (v4f*)out + i);
}

// ---------------------------------------------------------------------------
extern "C" void kernel_launch(void* const* d_in, const int* in_sizes, int n_in,
                              void* d_out, int out_size, void* d_ws, size_t ws_size,
                              hipStream_t stream) {
  const float* uemb = (const float*)d_in[0];   // N_USERS*64 f32
  const float* iemb = (const float*)d_in[1];   // N_ITEMS*64 f32
  const float* vals = (const float*)d_in[2];   // NNZ f32
  const int*   rows = (const int*)d_in[3];     // NNZ i32
  const int*   cols = (const int*)d_in[4];     // NNZ i32

  const long long nu_elems = in_sizes[0];
  const long long ni_elems = in_sizes[1];
  const long long nnz      = in_sizes[2];
  const long long n_elems  = nu_elems + ni_elems;  // N * 64 floats
  const long long n4       = n_elems / 4;
  const long long nu4      = nu_elems / 4;

  float* out  = (float*)d_out;
  float* bufA = (float*)d_ws;            // ping
  float* bufB = bufA + n_elems;          // pong (ws needs 2 * N*64*4 bytes)

  const int grid_e = (int)((n4 + TPB - 1) / TPB);
  const int grid_s = (int)((nnz + EPB - 1) / EPB);

  lgcn_init<<<grid_e, TPB, 0, stream>>>(uemb, iemb, nu4, n4, bufA, out, bufB);

  float* cur = bufA;
  float* nxt = bufB;
  for (int layer = 0; layer < 3; ++layer) {
    lgcn_spmm<<<grid_s, TPB, 0, stream>>>(rows, cols, vals, nnz, cur, nxt);
    if (layer < 2) {
      // accumulate this layer, zero `cur` (it becomes the next scatter target)
      lgcn_add_zero<<<grid_e, TPB, 0, stream>>>(out, nxt, cur, n4);
    } else {
      lgcn_add_scale<<<grid_e, TPB, 0, stream>>>(out, nxt, n4);
    }
    float* tmp = cur; cur = nxt; nxt = tmp;
  }
}